// GraphiT_Spectra_LSPE_Layer_17703855194488
// MI455X (gfx1250) — compile-verified
//
#include <hip/hip_runtime.h>
#include <hip/hip_bf16.h>

typedef __attribute__((ext_vector_type(2))) float v2f;
typedef __attribute__((ext_vector_type(8))) float v8f;

__device__ __forceinline__ v8f wmma4(v2f a, v2f b, v8f c) {
  // V_WMMA_F32_16X16X4_F32 : D = A(16x4) x B(4x16) + C(16x16)
  return __builtin_amdgcn_wmma_f32_16x16x4_f32(
      /*neg_a=*/false, a, /*neg_b=*/false, b,
      /*c_mod=*/(short)0, c, /*reuse_a=*/false, /*reuse_b=*/false);
}

// ---------------- degree / laplacian weights ----------------
__global__ void deg_kernel(const int* __restrict__ dst, float* __restrict__ deg, int E) {
  int tid = blockIdx.x * blockDim.x + threadIdx.x;
  if (tid < E) atomicAdd(&deg[dst[tid]], 1.0f);
}

__global__ void dinv_kernel(const float* __restrict__ deg, float* __restrict__ dinv, int N) {
  int tid = blockIdx.x * blockDim.x + threadIdx.x;
  if (tid < N) {
    float d = deg[tid];
    d = d < 1.0f ? 1.0f : d;
    dinv[tid] = rsqrtf(d);
  }
}

__global__ void wlap_kernel(const float* __restrict__ dinv, const int* __restrict__ src,
                            const int* __restrict__ dst, float* __restrict__ wlap, int E) {
  int tid = blockIdx.x * blockDim.x + threadIdx.x;
  if (tid < E) wlap[tid] = -(dinv[src[tid]] * dinv[dst[tid]]);
}

// out[dst,f] += w[e] * x[src,f]   (one thread per (edge, feature), D=64)
__global__ void scatter_wadd(const float* __restrict__ x, const float* __restrict__ w,
                             const int* __restrict__ src, const int* __restrict__ dst,
                             float* __restrict__ out, long long total) {
  long long tid = (long long)blockIdx.x * blockDim.x + threadIdx.x;
  if (tid >= total) return;
  int f = (int)(tid & 63);
  long long e = tid >> 6;
  int s = src[e], d = dst[e];
  float v = x[(size_t)s * 64 + f] * w[e];
  atomicAdd(&out[(size_t)d * 64 + f], v);
}

// out = 2*a - b  (Chebyshev recurrence)
__global__ void combine_kernel(const float* __restrict__ a, const float* __restrict__ b,
                               float* __restrict__ out, long long n) {
  long long tid = (long long)blockIdx.x * blockDim.x + threadIdx.x;
  if (tid < n) out[tid] = 2.0f * a[tid] - b[tid];
}

__global__ void concat_hp(const float* __restrict__ h, const float* __restrict__ p,
                          float* __restrict__ hp, int N) {
  long long tid = (long long)blockIdx.x * blockDim.x + threadIdx.x;
  if (tid >= (long long)N * 128) return;
  int n = (int)(tid >> 7);
  int c = (int)(tid & 127);
  hp[tid] = (c < 64) ? h[(size_t)n * 64 + c] : p[(size_t)n * 64 + (c - 64)];
}

// ---------------- generic WMMA GEMM: Out[rows x 64] = A[rows x kdim] @ B[kdim x 64] (+bias) ----
__global__ void gemm_wmma_kernel(const float* __restrict__ A, int lda,
                                 const float* __restrict__ B,
                                 const float* __restrict__ bias,
                                 float* __restrict__ Out, int ldo, int coloff,
                                 int rows, int kdim) {
  int wave = blockIdx.x * (blockDim.x >> 5) + (threadIdx.x >> 5);
  int rowTiles = (rows + 15) >> 4;
  if (wave >= rowTiles * 4) return;  // wave-uniform exit, EXEC stays full
  int rt = wave >> 2, ct = wave & 3;
  int lane = threadIdx.x & 31, half = lane >> 4, l16 = lane & 15;
  int ar = rt * 16 + l16;
  if (ar >= rows) ar = rows - 1;  // clamp, stores guarded below
  int bc = ct * 16 + l16;
  v8f acc = {};
  for (int k0 = 0; k0 < kdim; k0 += 4) {
    int kr = k0 + 2 * half;
    v2f a, b;
    a.x = A[(size_t)ar * lda + kr];
    a.y = A[(size_t)ar * lda + kr + 1];
    b.x = B[(size_t)kr * 64 + bc];
    b.y = B[(size_t)(kr + 1) * 64 + bc];
    acc = wmma4(a, b, acc);
  }
  float bv = bias ? bias[ct * 16 + l16] : 0.0f;
#pragma unroll
  for (int v = 0; v < 8; ++v) {
    int r = rt * 16 + v + 8 * half;
    if (r < rows) Out[(size_t)r * ldo + coloff + ct * 16 + l16] = acc[v] + bv;
  }
}

// ---------------- Chebyshev fused GEMM: Out[:,0:64] = sum_k (fc[k]*Txk) @ Wc[k] + b ----------
__global__ void cheb_gemm_kernel(const float* __restrict__ Tx0, const float* __restrict__ Tx1,
                                 const float* __restrict__ Tx2, const float* __restrict__ Tx3,
                                 const float* __restrict__ fc, const float* __restrict__ Wc,
                                 const float* __restrict__ bias, float* __restrict__ Out,
                                 int rows, int Nn) {
  int wave = blockIdx.x * (blockDim.x >> 5) + (threadIdx.x >> 5);
  int rowTiles = (rows + 15) >> 4;
  if (wave >= rowTiles * 4) return;
  int rt = wave >> 2, ct = wave & 3;
  int lane = threadIdx.x & 31, half = lane >> 4, l16 = lane & 15;
  int ar = rt * 16 + l16;
  if (ar >= rows) ar = rows - 1;
  int bc = ct * 16 + l16;
  const float* As[4] = {Tx0, Tx1, Tx2, Tx3};
  v8f acc = {};
#pragma unroll
  for (int k = 0; k < 4; ++k) {
    const float* A = As[k];
    const float* B = Wc + (size_t)k * 64 * 64;
    float sc = fc[(size_t)k * Nn + ar];  // per-row filter coefficient
#pragma unroll
    for (int k0 = 0; k0 < 64; k0 += 4) {
      int kr = k0 + 2 * half;
      v2f a, b;
      a.x = sc * A[(size_t)ar * 64 + kr];
      a.y = sc * A[(size_t)ar * 64 + kr + 1];
      b.x = B[(size_t)kr * 64 + bc];
      b.y = B[(size_t)(kr + 1) * 64 + bc];
      acc = wmma4(a, b, acc);
    }
  }
  float bv = bias[ct * 16 + l16];
#pragma unroll
  for (int v = 0; v < 8; ++v) {
    int r = rt * 16 + v + 8 * half;
    if (r < rows) Out[(size_t)r * 192 + ct * 16 + l16] = acc[v] + bv;
  }
}

// ---------------- fused attention: Eh = e@WE via WMMA -> LDS -> scores -> atomic scatter ------
// block = 128 threads = 4 waves; each wave owns one 16-edge tile.
__global__ void attn_kernel(const float* __restrict__ Ee, const float* __restrict__ WE,
                            const float* __restrict__ Qh, const float* __restrict__ Kh,
                            const float* __restrict__ Vh,
                            const int* __restrict__ src, const int* __restrict__ dst,
                            float* __restrict__ wV, float* __restrict__ z, int E) {
  __shared__ float lds[4 * 16 * 64];  // 16 KB: 16 edges x 64 cols per wave
  int wid = threadIdx.x >> 5;
  int tile = blockIdx.x * 4 + wid;
  int eTiles = (E + 15) >> 4;
  bool valid = tile < eTiles;
  int t = valid ? tile : 0;
  int e0 = t * 16;
  int lane = threadIdx.x & 31, half = lane >> 4, l16 = lane & 15;
  int ar = e0 + l16;
  if (ar >= E) ar = E - 1;
  float* myLds = lds + wid * 1024;

#pragma unroll
  for (int ct = 0; ct < 4; ++ct) {
    v8f acc = {};
#pragma unroll
    for (int k0 = 0; k0 < 64; k0 += 4) {
      int kr = k0 + 2 * half;
      v2f a, b;
      a.x = Ee[(size_t)ar * 64 + kr];
      a.y = Ee[(size_t)ar * 64 + kr + 1];
      int bc = ct * 16 + l16;
      b.x = WE[(size_t)kr * 64 + bc];
      b.y = WE[(size_t)(kr + 1) * 64 + bc];
      acc = wmma4(a, b, acc);
    }
#pragma unroll
    for (int v = 0; v < 8; ++v)
      myLds[(v + 8 * half) * 64 + ct * 16 + l16] = acc[v];
  }
  __syncthreads();
  if (!valid) return;  // wave-uniform

  const float scale = 0.35355339059327373f;  // 1/sqrt(DH=8)
#pragma unroll
  for (int pp = 0; pp < 4; ++pp) {
    int pair = pp * 32 + (threadIdx.x & 31);  // (edge m, head hh)
    int m = pair >> 3, hh = pair & 7;
    int em = e0 + m;
    if (em >= E) continue;
    int s = src[em], d = dst[em];
    float sc = 0.0f;
#pragma unroll
    for (int dd = 0; dd < 8; ++dd) {
      int c = hh * 8 + dd;
      sc += Kh[(size_t)s * 64 + c] * Qh[(size_t)d * 64 + c] * myLds[m * 64 + c];
    }
    sc *= scale;
    sc = fminf(5.0f, fmaxf(-5.0f, sc));
    float ss = __expf(sc);
    atomicAdd(&z[(size_t)d * 8 + hh], ss);
#pragma unroll
    for (int dd = 0; dd < 8; ++dd) {
      int c = hh * 8 + dd;
      atomicAdd(&wV[(size_t)d * 64 + c], Vh[(size_t)s * 64 + c] * ss);
    }
  }
}

__global__ void finalize_kernel(const float* __restrict__ wV, const float* __restrict__ z,
                                float* __restrict__ out, int N) {
  long long tid = (long long)blockIdx.x * blockDim.x + threadIdx.x;
  if (tid >= (long long)N * 64) return;
  int n = (int)(tid >> 6);
  int c = (int)(tid & 63);
  float zz = z[(size_t)n * 8 + (c >> 3)] + 1e-6f;
  out[(size_t)n * 192 + 128 + c] = wV[tid] / zz;
}

// ---------------- host-side orchestration ----------------
extern "C" void kernel_launch(void* const* d_in, const int* in_sizes, int n_in,
                              void* d_out, int out_size, void* d_ws, size_t ws_size,
                              hipStream_t stream) {
  const float* h  = (const float*)d_in[0];
  const float* p  = (const float*)d_in[1];
  const float* e  = (const float*)d_in[2];
  const float* ew = (const float*)d_in[3];
  const float* fc = (const float*)d_in[4];
  const int* esrc = (const int*)d_in[5];
  const int* edst = (const int*)d_in[6];
  const float* Wc = (const float*)d_in[7];
  const float* bc = (const float*)d_in[8];
  const float* Wg = (const float*)d_in[9];
  const float* bg = (const float*)d_in[10];
  const float* WQ = (const float*)d_in[11];
  const float* WK = (const float*)d_in[12];
  const float* WVm = (const float*)d_in[13];
  const float* WE = (const float*)d_in[14];
  float* out = (float*)d_out;

  const int N = in_sizes[0] / 64;
  const int E = in_sizes[5];
  const size_t ND = (size_t)N * 64;

  // workspace layout (floats)
  float* w = (float*)d_ws;
  float* deg  = w; w += N;
  float* dinv = w; w += N;
  float* wlap = w; w += E;
  float* Tx1  = w; w += ND;
  float* Tx2  = w; w += ND;
  float* Tx3  = w; w += ND;
  float* tmp  = w; w += ND;
  float* gcn  = w; w += ND;
  float* hp   = w; w += 2 * ND;
  float* Qh   = w; w += ND;
  float* Kh   = w; w += ND;
  float* Vh   = w; w += ND;
  float* wVb  = w; w += ND;
  float* zb   = w; w += (size_t)N * 8;

  hipMemsetAsync(deg, 0, sizeof(float) * N, stream);
  hipMemsetAsync(Tx1, 0, sizeof(float) * ND, stream);
  hipMemsetAsync(tmp, 0, sizeof(float) * ND, stream);
  hipMemsetAsync(gcn, 0, sizeof(float) * ND, stream);
  hipMemsetAsync(wVb, 0, sizeof(float) * ND, stream);
  hipMemsetAsync(zb, 0, sizeof(float) * (size_t)N * 8, stream);

  const int B = 256;
  const long long EF = (long long)E * 64;
  const int efBlocks = (int)((EF + B - 1) / B);
  const int ndBlocks = (int)(((long long)ND + B - 1) / B);

  // Laplacian edge weights
  deg_kernel<<<(E + B - 1) / B, B, 0, stream>>>(edst, deg, E);
  dinv_kernel<<<(N + B - 1) / B, B, 0, stream>>>(deg, dinv, N);
  wlap_kernel<<<(E + B - 1) / B, B, 0, stream>>>(dinv, esrc, edst, wlap, E);

  // Chebyshev recurrence: Tx1 = L~ h ; Tx2 = 2 L~ Tx1 - h ; Tx3 = 2 L~ Tx2 - Tx1
  scatter_wadd<<<efBlocks, B, 0, stream>>>(h, wlap, esrc, edst, Tx1, EF);
  scatter_wadd<<<efBlocks, B, 0, stream>>>(Tx1, wlap, esrc, edst, tmp, EF);
  combine_kernel<<<ndBlocks, B, 0, stream>>>(tmp, h, Tx2, (long long)ND);
  hipMemsetAsync(tmp, 0, sizeof(float) * ND, stream);
  scatter_wadd<<<efBlocks, B, 0, stream>>>(Tx2, wlap, esrc, edst, tmp, EF);
  combine_kernel<<<ndBlocks, B, 0, stream>>>(tmp, Tx1, Tx3, (long long)ND);

  // dense GEMMs (WMMA f32 16x16x4), 8 waves / 256-thread block
  const int rowTiles = (N + 15) / 16;
  const int gemmBlocks = (rowTiles * 4 + 7) / 8;
  cheb_gemm_kernel<<<gemmBlocks, 256, 0, stream>>>(h, Tx1, Tx2, Tx3, fc, Wc, bc, out, N, N);

  // GCN branch -> out[:,64:128]
  scatter_wadd<<<efBlocks, B, 0, stream>>>(h, ew, esrc, edst, gcn, EF);
  gemm_wmma_kernel<<<gemmBlocks, 256, 0, stream>>>(gcn, 64, Wg, bg, out, 192, 64, N, 64);

  // attention branch
  concat_hp<<<(int)(((long long)N * 128 + B - 1) / B), B, 0, stream>>>(h, p, hp, N);
  gemm_wmma_kernel<<<gemmBlocks, 256, 0, stream>>>(hp, 128, WQ, nullptr, Qh, 64, 0, N, 128);
  gemm_wmma_kernel<<<gemmBlocks, 256, 0, stream>>>(hp, 128, WK, nullptr, Kh, 64, 0, N, 128);
  gemm_wmma_kernel<<<gemmBlocks, 256, 0, stream>>>(hp, 128, WVm, nullptr, Vh, 64, 0, N, 128);

  const int eTiles = (E + 15) / 16;
  attn_kernel<<<(eTiles + 3) / 4, 128, 0, stream>>>(e, WE, Qh, Kh, Vh, esrc, edst, wVb, zb, E);
  finalize_kernel<<<(int)(((long long)N * 64 + B - 1) / B), B, 0, stream>>>(wVb, zb, out, N);
}